// NTXentLoss_viscy_19095424598749
// MI455X (gfx1250) — compile-verified
//
#include <hip/hip_runtime.h>
#include <hip/hip_bf16.h>
#include <math.h>

#define TWO_B 8192
#define HALF_B 4096
#define D 128
#define INV_T 2.0f          // 1 / 0.5 temperature
#define SMAX   2.0f         // static logsumexp shift: |cos|/T <= 2 for unit vectors

typedef __attribute__((ext_vector_type(2))) float v2f;
typedef __attribute__((ext_vector_type(8))) float v8f;

// ---------------------------------------------------------------------------
// Kernel 1: row-normalize embeddings (one wave32 per row, float4 vector loads)
// ---------------------------------------------------------------------------
__global__ __launch_bounds__(32) void ntx_normalize(const float* __restrict__ E,
                                                    float* __restrict__ En) {
    const int r = blockIdx.x;
    const int lane = threadIdx.x;                     // 0..31
    const float4* src = (const float4*)(E + (size_t)r * D);
    float4 v = src[lane];                             // 32 lanes * 4 = 128
    float ss = v.x * v.x + v.y * v.y + v.z * v.z + v.w * v.w;
#pragma unroll
    for (int off = 1; off < 32; off <<= 1)
        ss += __shfl_xor(ss, off, 32);
    const float inv = 1.0f / sqrtf(ss);               // norms ~11 >> eps
    float4 o;
    o.x = v.x * inv; o.y = v.y * inv; o.z = v.z * inv; o.w = v.w * inv;
    ((float4*)(En + (size_t)r * D))[lane] = o;
}

// ---------------------------------------------------------------------------
// Kernel 2: fused WMMA GEMM + masked static-shift logsumexp + picked logit.
// One block = one 32-row tile (two 16-row A tiles sharing every B fragment);
// 4 waves split the 512 column tiles round-robin. Nothing O(N^2) hits memory.
// ---------------------------------------------------------------------------
__global__ __launch_bounds__(128) void ntx_main(const float* __restrict__ En,
                                                const int* __restrict__ labels,
                                                float* __restrict__ rowloss) {
    const int mBase   = blockIdx.x * 32;
    const int wave    = threadIdx.x >> 5;             // 0..3
    const int lane    = threadIdx.x & 31;
    const int halfSel = lane >> 4;                    // selects K{0,1}/K{2,3} and row half
    const int lane16  = lane & 15;

    // Preload both A tiles for the whole K=128 (32 k-steps of 4): 128 VGPRs/lane.
    // ISA layout: A 16x4 f32 -> lane = M (0..15), VGPR0 = K{0|2}, VGPR1 = K{1|3}.
    const float* a0p = En + (size_t)(mBase + lane16) * D + 2 * halfSel;
    const float* a1p = a0p + 16 * D;
    v2f a0[32], a1[32];
#pragma unroll
    for (int k = 0; k < 32; ++k) {
        a0[k] = *(const v2f*)(a0p + 4 * k);
        a1[k] = *(const v2f*)(a1p + 4 * k);
    }

    // exp-sum state (static shift SMAX): 8 rows per lane per tile.
    float s0[8], s1[8];
#pragma unroll
    for (int j = 0; j < 8; ++j) { s0[j] = 0.0f; s1[j] = 0.0f; }

    // Column tiles that can contain masked (diagonal / positive-pair) entries.
    const int pBase = (mBase < HALF_B) ? mBase + HALF_B : mBase - HALF_B;
    const int tD = mBase >> 4;                        // diagonal tiles tD, tD+1
    const int tP = pBase >> 4;                        // partner tiles tP, tP+1

    for (int nt = wave; nt < TWO_B / 16; nt += 4) {
        const int nBase = nt * 16;
        // B 4x16 f32: lane = N, VGPR0 = K{0|2}, VGPR1 = K{1|3} -> same per-lane
        // address pattern as A since B = En^T tile of a symmetric Gram matrix.
        const float* bp = En + (size_t)(nBase + lane16) * D + 2 * halfSel;
        v8f c0 = {};
        v8f c1 = {};
#pragma unroll
        for (int k = 0; k < 32; ++k) {
            v2f b = *(const v2f*)(bp + 4 * k);
            c0 = __builtin_amdgcn_wmma_f32_16x16x4_f32(false, a0[k], false, b,
                                                       (short)0, c0, false, false);
            c1 = __builtin_amdgcn_wmma_f32_16x16x4_f32(false, a1[k], false, b,
                                                       (short)0, c1, false, false);
        }
        // C/D layout: lane = N (=nBase+lane16), VGPR j = row 8*halfSel + j.
        if (nt == tD || nt == tD + 1 || nt == tP || nt == tP + 1) {
            const int col = nBase + lane16;
#pragma unroll
            for (int j = 0; j < 8; ++j) {
                const int r0 = mBase + 8 * halfSel + j;
                const int r1 = r0 + 16;
                const int p0 = (r0 < HALF_B) ? r0 + HALF_B : r0 - HALF_B;
                const int p1 = (r1 < HALF_B) ? r1 + HALF_B : r1 - HALF_B;
                float g0 = fmaf(INV_T, c0[j], -SMAX);
                float g1 = fmaf(INV_T, c1[j], -SMAX);
                if (col == r0 || col == p0) g0 = -INFINITY;     // exp -> 0
                if (col == r1 || col == p1) g1 = -INFINITY;
                s0[j] += __expf(g0);
                s1[j] += __expf(g1);
            }
        } else {
#pragma unroll
            for (int j = 0; j < 8; ++j) {
                s0[j] += __expf(fmaf(INV_T, c0[j], -SMAX));
                s1[j] += __expf(fmaf(INV_T, c1[j], -SMAX));
            }
        }
    }

    // Sum across the 16 lanes of each half-wave (same row set, disjoint cols).
#pragma unroll
    for (int off = 1; off < 16; off <<= 1) {
#pragma unroll
        for (int j = 0; j < 8; ++j) {
            s0[j] += __shfl_xor(s0[j], off, 32);
            s1[j] += __shfl_xor(s1[j], off, 32);
        }
    }

    // Combine the 4 waves' partial sums through LDS.
    __shared__ float ls[4][32];
    if (lane16 == 0) {
#pragma unroll
        for (int j = 0; j < 8; ++j) {
            ls[wave][8 * halfSel + j]      = s0[j];
            ls[wave][16 + 8 * halfSel + j] = s1[j];
        }
    }
    __syncthreads();

    if (threadIdx.x < 32) {
        const int t = threadIdx.x;
        const float S = ls[0][t] + ls[1][t] + ls[2][t] + ls[3][t];
        const float logz = SMAX + __logf(S);

        // Picked logit: map label index in the masked [2B-2] row to a column.
        const int r  = mBase + t;
        const int p  = (r < HALF_B) ? r + HALF_B : r - HALF_B;
        const int m1 = (r < p) ? r : p;
        const int m2 = (r < p) ? p : r;
        int ccol = labels[r];
        if (ccol >= m1) ccol += 1;
        if (ccol >= m2) ccol += 1;

        const float4* pr = (const float4*)(En + (size_t)r * D);
        const float4* pc = (const float4*)(En + (size_t)ccol * D);
        float dot = 0.0f;
#pragma unroll
        for (int k = 0; k < D / 4; ++k) {
            const float4 x = pr[k], y = pc[k];
            dot += x.x * y.x + x.y * y.y + x.z * y.z + x.w * y.w;
        }
        rowloss[r] = logz - dot * INV_T;
    }
}

// ---------------------------------------------------------------------------
// Kernel 3: deterministic single-block tree reduction -> scalar loss.
// ---------------------------------------------------------------------------
__global__ __launch_bounds__(256) void ntx_reduce(const float* __restrict__ rowloss,
                                                  float* __restrict__ out) {
    __shared__ float sm[256];
    float acc = 0.0f;
    for (int i = threadIdx.x; i < TWO_B; i += 256)
        acc += rowloss[i];
    sm[threadIdx.x] = acc;
    __syncthreads();
#pragma unroll
    for (int off = 128; off > 0; off >>= 1) {
        if ((int)threadIdx.x < off) sm[threadIdx.x] += sm[threadIdx.x + off];
        __syncthreads();
    }
    if (threadIdx.x == 0) out[0] = sm[0] / (float)TWO_B;
}

extern "C" void kernel_launch(void* const* d_in, const int* in_sizes, int n_in,
                              void* d_out, int out_size, void* d_ws, size_t ws_size,
                              hipStream_t stream) {
    (void)in_sizes; (void)n_in; (void)out_size; (void)ws_size;
    const float* E      = (const float*)d_in[0];   // [8192, 128] f32
    const int*   labels = (const int*)d_in[1];     // [8192] i32
    float* out     = (float*)d_out;                // scalar f32
    float* En      = (float*)d_ws;                 // 8192*128 f32 = 4 MB
    float* rowloss = En + (size_t)TWO_B * D;       // 8192 f32

    ntx_normalize<<<TWO_B, 32, 0, stream>>>(E, En);
    ntx_main<<<TWO_B / 32, 128, 0, stream>>>(En, labels, rowloss);
    ntx_reduce<<<1, 256, 0, stream>>>(rowloss, out);
}